// MipRayMarcher2_51651276701982
// MI455X (gfx1250) — compile-verified
//
#include <hip/hip_runtime.h>
#include <math.h>

typedef float v2f __attribute__((ext_vector_type(2)));
typedef float v8f __attribute__((ext_vector_type(8)));

#define NS   96   // samples per ray
#define NSM1 95   // midpoints per ray
#define NC   32   // channels

// ---------------------------------------------------------------------------
// Kernel 0: init min/max accumulators in workspace (ws is poisoned by harness)
// ---------------------------------------------------------------------------
__global__ void mrm_init_minmax(int* __restrict__ wsI) {
    if (threadIdx.x == 0) {
        wsI[0] = 0x7F800000;  // +inf bits (running min, depths > 0)
        wsI[1] = 0;           // 0 bits    (running max, depths > 0)
    }
}

// ---------------------------------------------------------------------------
// Kernel 1: global depth min/max. depths sorted ascending along S, so
// per-ray min = depths[ray][0], per-ray max = depths[ray][S-1].
// All depths > 0  =>  int bit-pattern ordering == float ordering.
// ---------------------------------------------------------------------------
__global__ void mrm_depth_minmax(const float* __restrict__ depths,
                                 int* __restrict__ wsI, int nrays) {
    int i = blockIdx.x * blockDim.x + threadIdx.x;
    int vmin = 0x7F800000;
    int vmax = 0;
    if (i < nrays) {
        vmin = __float_as_int(depths[(size_t)i * NS + 0]);
        vmax = __float_as_int(depths[(size_t)i * NS + (NS - 1)]);
    }
    #pragma unroll
    for (int off = 16; off > 0; off >>= 1) {
        vmin = min(vmin, __shfl_xor(vmin, off, 32));
        vmax = max(vmax, __shfl_xor(vmax, off, 32));
    }
    if ((threadIdx.x & 31) == 0) {
        atomicMin(&wsI[0], vmin);
        atomicMax(&wsI[1], vmax);
    }
}

// ---------------------------------------------------------------------------
// Kernel 2: main ray march. One wave32 per ray, lane = channel.
// 3 slot-groups x 8 chunks x 4 samples (= 96 padded samples).
// Per chunk: 4 coalesced non-temporal 128B color-row loads, uniform scan
// with fast transcendentals, weighted color sum via two
// V_WMMA_F32_16X16X4_F32 accumulated in C/D across the whole ray.
// weights/alpha gathered into per-lane registers, stored coalesced per slot.
// ---------------------------------------------------------------------------
__global__ __launch_bounds__(256) void mrm_raymarch(
    const float* __restrict__ colors,
    const float* __restrict__ densities,
    const float* __restrict__ depths,
    float* __restrict__ out_rgb,     // [nrays, 32]
    float* __restrict__ out_depth,   // [nrays]
    float* __restrict__ out_w,       // [nrays, 95]
    float* __restrict__ out_a,       // [nrays, 95]
    float* __restrict__ out_bg,      // [nrays]
    const int* __restrict__ wsI,
    int nrays)
{
    const int lane = threadIdx.x & 31;
    const int ray  = blockIdx.x * (blockDim.x >> 5) + (threadIdx.x >> 5);
    if (ray >= nrays) return;   // wave-uniform: EXEC stays all-ones for WMMA

    const float* cbase = colors    + (size_t)ray * NS * NC;
    const float* dbase = densities + (size_t)ray * NS;
    const float* zbase = depths    + (size_t)ray * NS;

    const float gmin = __int_as_float(wsI[0]);
    const float gmax = __int_as_float(wsI[1]);

    v8f acc1 = {0.f,0.f,0.f,0.f,0.f,0.f,0.f,0.f};   // channels  0..15
    v8f acc2 = {0.f,0.f,0.f,0.f,0.f,0.f,0.f,0.f};   // channels 16..31

    float T = 1.0f;          // transmittance (uniform across wave)
    float dep_acc = 0.0f;    // sum w * depth_mid
    float wt      = 0.0f;    // sum w

    float row0 = __builtin_nontemporal_load(cbase + lane);   // carried row s
    float dprv = dbase[0];
    float zprv = zbase[0];

    for (int slot = 0; slot < 3; ++slot) {
        float wcur = 0.0f;   // this lane's weight  for sample slot*32+lane
        float acur = 0.0f;   // this lane's alpha   for sample slot*32+lane

        for (int j = 0; j < 8; ++j) {
            const int s  = slot * 32 + j * 4;
            const int t1 = min(s + 1, NS - 1);
            const int t2 = min(s + 2, NS - 1);
            const int t3 = min(s + 3, NS - 1);
            const int t4 = min(s + 4, NS - 1);

            // coalesced 128B row loads, streamed once -> non-temporal
            float r1 = __builtin_nontemporal_load(cbase + (size_t)t1 * NC + lane);
            float r2 = __builtin_nontemporal_load(cbase + (size_t)t2 * NC + lane);
            float r3 = __builtin_nontemporal_load(cbase + (size_t)t3 * NC + lane);
            float r4 = __builtin_nontemporal_load(cbase + (size_t)t4 * NC + lane);
            if (s + 8 < NS)
                __builtin_prefetch(cbase + (size_t)(s + 8) * NC + lane, 0, 0);

            float dd0 = dprv, dd1 = dbase[t1], dd2 = dbase[t2], dd3 = dbase[t3], dd4 = dbase[t4];
            float zz0 = zprv, zz1 = zbase[t1], zz2 = zbase[t2], zz3 = zbase[t3], zz4 = zbase[t4];

            // midpoint color rows (lane = channel)
            float cm0 = 0.5f * (row0 + r1);
            float cm1 = 0.5f * (r1 + r2);
            float cm2 = 0.5f * (r2 + r3);
            float cm3 = 0.5f * (r3 + r4);

            // ---- uniform sequential scan over up to 4 samples -------------
            float w[4];
            float dA[5] = {dd0, dd1, dd2, dd3, dd4};
            float zA[5] = {zz0, zz1, zz2, zz3, zz4};
            #pragma unroll
            for (int k = 0; k < 4; ++k) {
                if (s + k < NSM1) {
                    float delta = zA[k + 1] - zA[k];
                    float dm    = 0.5f * (dA[k] + dA[k + 1]) - 1.0f;
                    // stable softplus, fast transcendental path
                    float sp    = fmaxf(dm, 0.0f) + __logf(1.0f + __expf(-fabsf(dm)));
                    float a     = 1.0f - __expf(-sp * delta);
                    w[k]  = a * T;
                    T    *= (1.0f - a + 1e-10f);
                    float zm = 0.5f * (zA[k] + zA[k + 1]);
                    dep_acc += w[k] * zm;
                    wt      += w[k];
                    // capture into the owning lane (slot uniform per chunk)
                    bool take = (lane == j * 4 + k);
                    wcur = take ? w[k] : wcur;
                    acur = take ? a    : acur;
                } else {
                    w[k] = 0.0f;
                }
            }

            // ---- A matrix 16x4 f32: only row M=0 nonzero ------------------
            // layout: VGPR0 = K0(lanes0-15)/K2(lanes16-31); VGPR1 = K1/K3
            v2f A;
            A.x = (lane == 0) ? w[0] : (lane == 16) ? w[2] : 0.0f;
            A.y = (lane == 0) ? w[1] : (lane == 16) ? w[3] : 0.0f;

            // ---- B matrices 4x16 f32 via half-wave swaps ------------------
            float sw0 = __shfl_xor(cm0, 16, 32);
            float sw1 = __shfl_xor(cm1, 16, 32);
            float sw2 = __shfl_xor(cm2, 16, 32);
            float sw3 = __shfl_xor(cm3, 16, 32);
            bool lo = (lane < 16);
            v2f B1, B2;
            B1.x = lo ? cm0 : sw2;   // K0 (lanes0-15), K2 (lanes16-31), ch 0..15
            B1.y = lo ? cm1 : sw3;   // K1, K3
            B2.x = lo ? sw0 : cm2;   // same, ch 16..31
            B2.y = lo ? sw1 : cm3;

            acc1 = __builtin_amdgcn_wmma_f32_16x16x4_f32(false, A, false, B1,
                                                         (short)0, acc1, false, false);
            acc2 = __builtin_amdgcn_wmma_f32_16x16x4_f32(false, A, false, B2,
                                                         (short)0, acc2, false, false);

            row0 = r4; dprv = dd4; zprv = zz4;
        }

        // ---- coalesced 128B weight/alpha stores for this slot -------------
        const int idx = slot * 32 + lane;
        if (idx < NSM1) {
            __builtin_nontemporal_store(wcur, out_w + (size_t)ray * NSM1 + idx);
            __builtin_nontemporal_store(acur, out_a + (size_t)ray * NSM1 + idx);
        }
    }

    // ---- finalize: D row M=0 lives in VGPR0, lanes 0..15 ------------------
    float hi = __shfl_xor(acc2[0], 16, 32);           // move ch16..31 to lanes 16..31
    float rgbv = (lane < 16) ? acc1[0] : hi;
    __builtin_nontemporal_store(rgbv * 2.0f - 1.0f,
                                out_rgb + (size_t)ray * NC + lane);  // coalesced

    if (lane == 0) {
        float cd = dep_acc / wt;
        if (cd != cd) cd = INFINITY;                  // NaN -> inf (then clipped)
        cd = fminf(fmaxf(cd, gmin), gmax);
        out_depth[ray] = cd;
        out_bg[ray]    = T;
    }
}

// ---------------------------------------------------------------------------
extern "C" void kernel_launch(void* const* d_in, const int* in_sizes, int n_in,
                              void* d_out, int out_size, void* d_ws, size_t ws_size,
                              hipStream_t stream) {
    const float* colors    = (const float*)d_in[0];
    const float* densities = (const float*)d_in[1];
    const float* depths    = (const float*)d_in[2];

    const int nrays = in_sizes[1] / NS;  // densities flat size = B*R*S

    float* out       = (float*)d_out;
    float* out_rgb   = out;                                   // [nrays,32]
    float* out_depth = out_rgb   + (size_t)nrays * NC;        // [nrays]
    float* out_w     = out_depth + (size_t)nrays;             // [nrays,95]
    float* out_a     = out_w     + (size_t)nrays * NSM1;      // [nrays,95]
    float* out_bg    = out_a     + (size_t)nrays * NSM1;      // [nrays]

    int* wsI = (int*)d_ws;

    mrm_init_minmax<<<1, 64, 0, stream>>>(wsI);
    mrm_depth_minmax<<<(nrays + 255) / 256, 256, 0, stream>>>(depths, wsI, nrays);

    const int waves_per_block = 8;                       // 256 threads
    const int blocks = (nrays + waves_per_block - 1) / waves_per_block;
    mrm_raymarch<<<blocks, 256, 0, stream>>>(colors, densities, depths,
                                             out_rgb, out_depth, out_w, out_a, out_bg,
                                             wsI, nrays);
}